// SphericalExpansion_21784074125332
// MI455X (gfx1250) — compile-verified
//
#include <hip/hip_runtime.h>
#include <hip/hip_bf16.h>

// CDNA5 / gfx1250: wave32, WMMA f32 16x16x4.
typedef float v2f __attribute__((ext_vector_type(2)));
typedef float v8f __attribute__((ext_vector_type(8)));

#define NTILES 18  // N-tiles of 16 over (n_l*4) = {84,72,60,48} -> 6+5+4+3

__global__ void se_zero_kernel(int* cnt, int A) {
    int t = blockIdx.x * blockDim.x + threadIdx.x;
    if (t < A) cnt[t] = 0;
}

__global__ void se_count_kernel(const int* __restrict__ idx, int* cnt, int P) {
    int p = blockIdx.x * blockDim.x + threadIdx.x;
    if (p < P) atomicAdd(&cnt[idx[p]], 1);
}

// Single-block exclusive scan over A<=8192 counts (1024 threads, 8 items each).
__global__ void se_scan_kernel(const int* __restrict__ cnt, int* start, int* cursor, int A) {
    __shared__ int sums[1024];
    const int t = threadIdx.x;
    const int C = (A + 1023) / 1024;
    const int base = t * C;
    int s = 0;
    for (int j = 0; j < C; ++j) {
        int k = base + j;
        if (k < A) s += cnt[k];
    }
    sums[t] = s;
    __syncthreads();
    for (int off = 1; off < 1024; off <<= 1) {
        int v = (t >= off) ? sums[t - off] : 0;
        __syncthreads();
        sums[t] += v;
        __syncthreads();
    }
    int run = (t == 0) ? 0 : sums[t - 1];
    for (int j = 0; j < C; ++j) {
        int k = base + j;
        if (k < A) {
            start[k] = run;
            cursor[k] = run;
            run += cnt[k];
        }
    }
}

__global__ void se_scatter_kernel(const int* __restrict__ idx, int* cursor,
                                  int* perm, int P) {
    int p = blockIdx.x * blockDim.x + threadIdx.x;
    if (p < P) {
        int a = idx[p];
        int pos = atomicAdd(&cursor[a], 1);
        perm[pos] = p;
    }
}

// ---------------- per-atom GEMM via V_WMMA_F32_16X16X4_F32 ----------------
// out_l[a][m][n][s] = sum_{p in atom a} Y[p][m] * (Sn[p][n] * fcut*(d/5)^l * W[z][s])
// A-matrix (16x4 f32): M = concatenated angular index (16 = 1+3+5+7),
//                      K = 4 edges per issue.
// B-matrix (4x16 f32): N = 16-wide tile of the flattened (n,s) dim per l.
// Key factorization: within a tile column, si = m&3 (tile-independent) and the
// Sn index is 4t + (m>>2) (l-independent), so each lane only needs 6 Sn values
// and 4 Rls values per edge; the 18 B operands are their cross products.
__launch_bounds__(32)
__global__ void se_expand_kernel(const float* __restrict__ R,
                                 const int* __restrict__ Zj,
                                 const float* __restrict__ W,
                                 const int* __restrict__ perm,
                                 const int* __restrict__ start,
                                 const int* __restrict__ cnt,
                                 float* __restrict__ out) {
    constexpr int NT_L[4]   = {6, 5, 4, 3};     // N tiles per l
    constexpr int NS4[4]    = {84, 72, 60, 48}; // n_l * 4
    constexpr int BLKOFF[4] = {0, 84, 300, 600};
    constexpr int M0[4]     = {0, 1, 4, 9};     // angular row offset of block l
    constexpr int NM[4]     = {1, 3, 5, 7};     // 2l+1

    const int a    = blockIdx.x;
    const int lane = threadIdx.x;

    __shared__ float Ys[32][16];   // angular values per staged edge
    __shared__ float Sn[32][24];   // sin(n pi d/5)/(arg+1e-6), n=1..21; [21..23]=0
    __shared__ float Rls[32][16];  // fcut*(d/5)^l * W[z][s], index l*4+s

    v8f acc[NTILES];
#pragma unroll
    for (int t = 0; t < NTILES; ++t)
        acc[t] = v8f{0.f, 0.f, 0.f, 0.f, 0.f, 0.f, 0.f, 0.f};

    const int beg = start[a];
    const int n_e = cnt[a];

    const int m     = lane & 15;
    const int khalf = (lane >> 4) * 2;  // lanes 0-15 -> K0/K1, 16-31 -> K2/K3
    const int nrow  = m >> 2;           // Sn sub-index from lane
    const int scol  = m & 3;            // species sub-index from lane
    const bool ok14 = (m < 8);          // partial tile (l=1, t=4)
    const bool ok23 = (m < 12);         // partial tile (l=2, t=3)

    for (int base = 0; base < n_e; base += 32) {
        // ---- stage up to 32 edges: one edge per lane ----
        {
            float Y[16], s[21], rls[16];
            const int e = base + lane;
            if (e < n_e) {
                const int p = perm[beg + e];
                const float x = R[3 * p + 0];
                const float y = R[3 * p + 1];
                const float z = R[3 * p + 2];
                const float d = sqrtf(x * x + y * y + z * z + 1e-12f);
                const float inv = 1.0f / d;
                const float ux = x * inv, uy = y * inv, uz = z * inv;
                const float x2 = ux * ux, y2 = uy * uy, z2 = uz * uz;
                Y[0]  = 0.28209479177387814f;
                Y[1]  = 0.4886025119029199f * uy;
                Y[2]  = 0.4886025119029199f * uz;
                Y[3]  = 0.4886025119029199f * ux;
                Y[4]  = 1.0925484305920792f * ux * uy;
                Y[5]  = 1.0925484305920792f * uy * uz;
                Y[6]  = 0.31539156525252005f * (3.0f * z2 - 1.0f);
                Y[7]  = 1.0925484305920792f * ux * uz;
                Y[8]  = 0.5462742152960396f * (x2 - y2);
                Y[9]  = 0.5900435899266435f * uy * (3.0f * x2 - y2);
                Y[10] = 2.890611442640554f * ux * uy * uz;
                Y[11] = 0.4570457994644658f * uy * (5.0f * z2 - 1.0f);
                Y[12] = 0.3731763325901154f * uz * (5.0f * z2 - 3.0f);
                Y[13] = 0.4570457994644658f * ux * (5.0f * z2 - 1.0f);
                Y[14] = 1.445305721320277f * uz * (x2 - y2);
                Y[15] = 0.5900435899266435f * ux * (x2 - 3.0f * y2);
                const float dc   = fminf(d, 5.0f);
                const float fcut = 0.5f * (__cosf(0.6283185307179586f * dc) + 1.0f);
                const float tt   = d * 0.2f;
                float pl[4];
                pl[0] = fcut;
                pl[1] = fcut * tt;
                pl[2] = pl[1] * tt;
                pl[3] = pl[2] * tt;
#pragma unroll
                for (int n = 1; n <= 21; ++n) {
                    const float arg = d * (0.6283185307179586f * (float)n);
                    s[n - 1] = __sinf(arg) / (arg + 1e-6f);
                }
                const int zi = Zj[p];
#pragma unroll
                for (int l = 0; l < 4; ++l)
#pragma unroll
                    for (int si = 0; si < 4; ++si)
                        rls[l * 4 + si] = pl[l] * W[zi * 4 + si];
            } else {
#pragma unroll
                for (int j = 0; j < 16; ++j) Y[j] = 0.f;
#pragma unroll
                for (int j = 0; j < 21; ++j) s[j] = 0.f;
#pragma unroll
                for (int j = 0; j < 16; ++j) rls[j] = 0.f;
            }
#pragma unroll
            for (int j = 0; j < 16; ++j) Ys[lane][j] = Y[j];
#pragma unroll
            for (int j = 0; j < 21; ++j) Sn[lane][j] = s[j];
            Sn[lane][21] = 0.f;   // padding: makes l=0 partial tile self-masking
            Sn[lane][22] = 0.f;
            Sn[lane][23] = 0.f;
#pragma unroll
            for (int j = 0; j < 16; ++j) Rls[lane][j] = rls[j];
        }
        __syncthreads();  // single-wave WG: compiles to LDS counter wait

        const int kmax    = n_e - base < 32 ? n_e - base : 32;
        const int ngroups = (kmax + 3) >> 2;

        for (int kg = 0; kg < ngroups; ++kg) {
            const int k0 = kg * 4 + khalf;
            v2f Am;
            Am.x = Ys[k0][m];      // A VGPR0 : K = khalf
            Am.y = Ys[k0 + 1][m];  // A VGPR1 : K = khalf+1

            // Unconditional gathers: 6 Sn + 4 Rls values per edge.
            float snA[6], snB[6], rlA[4], rlB[4];
#pragma unroll
            for (int t = 0; t < 6; ++t) {
                snA[t] = Sn[k0][4 * t + nrow];
                snB[t] = Sn[k0 + 1][4 * t + nrow];
            }
#pragma unroll
            for (int l = 0; l < 4; ++l) {
                rlA[l] = Rls[k0][l * 4 + scol];
                rlB[l] = Rls[k0 + 1][l * 4 + scol];
            }

            int tile = 0;
#pragma unroll
            for (int l = 0; l < 4; ++l) {
#pragma unroll
                for (int t = 0; t < NT_L[l]; ++t) {
                    v2f Bm;
                    Bm.x = snA[t] * rlA[l];
                    Bm.y = snB[t] * rlB[l];
                    if (l == 1 && t == 4) {  // valid only for m < 8
                        Bm.x = ok14 ? Bm.x : 0.f;
                        Bm.y = ok14 ? Bm.y : 0.f;
                    }
                    if (l == 2 && t == 3) {  // valid only for m < 12
                        Bm.x = ok23 ? Bm.x : 0.f;
                        Bm.y = ok23 ? Bm.y : 0.f;
                    }
                    acc[tile] = __builtin_amdgcn_wmma_f32_16x16x4_f32(
                        false, Am, false, Bm, (short)0, acc[tile], false, false);
                    ++tile;
                }
            }
        }
        __syncthreads();  // staging buffers reused next iteration
    }

    // ---- store: D VGPR v holds rows M=v (lanes 0-15) / M=v+8 (lanes 16-31) ----
    const int mrow = (lane >> 4) * 8;
    float* outa = out + (size_t)a * 936;
    int tile = 0;
#pragma unroll
    for (int l = 0; l < 4; ++l) {
#pragma unroll
        for (int t = 0; t < NT_L[l]; ++t) {
            const int ng = t * 16 + m;
            if (ng < NS4[l]) {
#pragma unroll
                for (int v = 0; v < 8; ++v) {
                    const int mr = v + mrow;      // global angular row
                    const int ml = mr - M0[l];    // row within block l
                    if (ml >= 0 && ml < NM[l])
                        outa[BLKOFF[l] + ml * NS4[l] + ng] = acc[tile][v];
                }
            }
            ++tile;
        }
    }
}

extern "C" void kernel_launch(void* const* d_in, const int* in_sizes, int n_in,
                              void* d_out, int out_size, void* d_ws, size_t ws_size,
                              hipStream_t stream) {
    const float* R = (const float*)d_in[0];   // (P,3) f32
    const int*   I = (const int*)d_in[1];     // (P,)  i32
    const int*   Z = (const int*)d_in[2];     // (P,)  i32
    const float* W = (const float*)d_in[3];   // (100,4) f32
    const int P = in_sizes[1];
    const int A = out_size / 936;             // 936 features per atom

    int* cnt    = (int*)d_ws;
    int* start  = cnt + A;
    int* cursor = start + A;
    int* perm   = cursor + A;

    se_zero_kernel<<<(A + 255) / 256, 256, 0, stream>>>(cnt, A);
    se_count_kernel<<<(P + 255) / 256, 256, 0, stream>>>(I, cnt, P);
    se_scan_kernel<<<1, 1024, 0, stream>>>(cnt, start, cursor, A);
    se_scatter_kernel<<<(P + 255) / 256, 256, 0, stream>>>(I, cursor, perm, P);
    se_expand_kernel<<<A, 32, 0, stream>>>(R, Z, W, perm, start, cnt, (float*)d_out);
}